// StochLoss_56470230007919
// MI455X (gfx1250) — compile-verified
//
#include <hip/hip_runtime.h>

// StochLoss on MI455X (gfx1250, wave32).
// Closed-form: loss = (S2 - S1^2/T) / (T-1), where
//   S1 = sum_i w_i * e_i,  S2 = sum_i w_i * e_i^2,  e = (y - y_hat)^2,
//   w depends only on the N index (w = n+1),  T = B * n(n+1)/2 * D.
// Memory-bound: 4 MiB read total -> ~0.18us at 23.3 TB/s. Streaming phase is
// float4 loads + VALU FMA; the cross-lane reduction uses V_WMMA_F32_16X16X4_F32
// (A = 64 lane partials in 2 VGPRs, B = 0/1 selector) to reduce BOTH moments
// across the wave in one matrix op at full f32 precision.

typedef float v2f __attribute__((ext_vector_type(2)));
typedef float v8f __attribute__((ext_vector_type(8)));

#define NB 8
#define NN 1024
#define ND 64
#define TOTAL  (NB * NN * ND)   // 524288 fp32 elements
#define TOTAL4 (TOTAL / 4)      // 131072 float4 vectors
#define GRID1  256
#define BLOCK  256              // 8 waves of 32

// Reduce (s2, s4) across all 32 lanes of the wave with one WMMA.
// A(16x4): lane L<16 holds A[L][0..1]=(s2,s4); lane L>=16 holds A[L-16][2..3].
// B(4x16): B[k][n] = 1 for (k even, n<8) or (k odd, n>=8), else 0.
//   => D[m][n<8]  = s2[m] + s2[m+16]
//      D[m][n>=8] = s4[m] + s4[m+16]
// Sum the 8 D VGPRs per lane, fold lane halves with shfl_xor(16):
//   lanes with (lane%16)<8  hold total S2, (lane%16)>=8 hold total S4.
// EXEC must be all ones here (callers guarantee no divergence).
__device__ __forceinline__ void wave_reduce2(float s2, float s4,
                                             float& o2, float& o4) {
    const int lane = threadIdx.x & 31;
    const int col  = lane & 15;
    v2f a; a[0] = s2; a[1] = s4;
    v2f b;
    b[0] = (col < 8) ? 1.0f : 0.0f;   // K even rows of B
    b[1] = (col < 8) ? 0.0f : 1.0f;   // K odd  rows of B
    v8f c = {};
    v8f d = __builtin_amdgcn_wmma_f32_16x16x4_f32(
        /*neg_a=*/false, a, /*neg_b=*/false, b,
        /*c_mod=*/(short)0, c, /*reuse_a=*/false, /*reuse_b=*/false);
    float t = ((d[0] + d[1]) + (d[2] + d[3])) +
              ((d[4] + d[5]) + (d[6] + d[7]));
    t += __shfl_xor(t, 16, 32);       // combine M=0..7 with M=8..15 halves
    o2 = __shfl(t, 0, 32);            // any lane with col<8
    o4 = __shfl(t, 8, 32);            // any lane with col>=8
}

__global__ void __launch_bounds__(BLOCK)
stoch_partial(const float4* __restrict__ y, const float4* __restrict__ yh,
              float* __restrict__ ws) {
    float s2 = 0.0f, s4 = 0.0f;
    const int nthreads = GRID1 * BLOCK;            // 65536; TOTAL4 % nthreads == 0
    for (int q = blockIdx.x * BLOCK + threadIdx.x; q < TOTAL4; q += nthreads) {
        float4 a = y[q];
        float4 b = yh[q];
        // flat elem index i = 4q; n = (i/64) % 1024 = (q/16) % 1024; w = n+1.
        // D=64 is a multiple of 4, so all 4 lanes of the float4 share one w.
        float w = (float)(((q >> 4) & (NN - 1)) + 1);
        float dx = a.x - b.x, dy = a.y - b.y, dz = a.z - b.z, dw = a.w - b.w;
        float ex = dx * dx, ey = dy * dy, ez = dz * dz, ew = dw * dw;
        s2 = fmaf(w, (ex + ey) + (ez + ew), s2);
        s4 = fmaf(w, (ex * ex + ey * ey) + (ez * ez + ew * ew), s4);
    }

    float w2, w4;
    wave_reduce2(s2, s4, w2, w4);                  // uniform control flow -> EXEC all 1s

    __shared__ float l2[BLOCK / 32], l4[BLOCK / 32];
    const int wid = threadIdx.x >> 5;
    if ((threadIdx.x & 31) == 0) { l2[wid] = w2; l4[wid] = w4; }
    __syncthreads();
    if (threadIdx.x == 0) {
        float t2 = 0.0f, t4 = 0.0f;
#pragma unroll
        for (int i = 0; i < BLOCK / 32; ++i) { t2 += l2[i]; t4 += l4[i]; }
        ws[blockIdx.x]         = t2;               // S1 partials in ws[0..255]
        ws[GRID1 + blockIdx.x] = t4;               // S2 partials in ws[256..511]
    }
}

__global__ void __launch_bounds__(BLOCK)
stoch_final(const float* __restrict__ ws, float* __restrict__ out) {
    // One block of 256 threads; one partial pair per thread (GRID1 == BLOCK).
    float s2 = ws[threadIdx.x];
    float s4 = ws[GRID1 + threadIdx.x];

    float w2, w4;
    wave_reduce2(s2, s4, w2, w4);                  // EXEC all 1s

    __shared__ float l2[BLOCK / 32], l4[BLOCK / 32];
    const int wid = threadIdx.x >> 5;
    if ((threadIdx.x & 31) == 0) { l2[wid] = w2; l4[wid] = w4; }
    __syncthreads();
    if (threadIdx.x == 0) {
        double t2 = 0.0, t4 = 0.0;
#pragma unroll
        for (int i = 0; i < BLOCK / 32; ++i) { t2 += (double)l2[i]; t4 += (double)l4[i]; }
        const double T = 8.0 * (1024.0 * 1025.0 * 0.5) * 64.0;  // 268,697,600
        const double var = (t4 - (t2 * t2) / T) / (T - 1.0);    // unbiased (ddof=1)
        out[0] = (float)var;
    }
}

extern "C" void kernel_launch(void* const* d_in, const int* in_sizes, int n_in,
                              void* d_out, int out_size, void* d_ws, size_t ws_size,
                              hipStream_t stream) {
    const float4* y  = (const float4*)d_in[0];
    const float4* yh = (const float4*)d_in[1];
    float* ws  = (float*)d_ws;   // uses 512 floats = 2 KiB
    float* out = (float*)d_out;  // scalar fp32 loss

    stoch_partial<<<GRID1, BLOCK, 0, stream>>>(y, yh, ws);
    stoch_final<<<1, BLOCK, 0, stream>>>(ws, out);
}